// OutputLayer_17695265259693
// MI455X (gfx1250) — compile-verified
//
#include <hip/hip_runtime.h>
#include <math.h>

typedef __attribute__((ext_vector_type(2))) float v2f;
typedef __attribute__((ext_vector_type(8))) float v8f;

// Problem constants (from reference)
constexpr int N_   = 4096;
constexpr int Q_   = 8;
constexpr int LT_  = 16;
constexpr int M_   = 128;
constexpr int DIN  = 256;       // 32*Q
constexpr int NH   = N_ - LT_;  // 4080
constexpr int K2   = 512;       // stacked-K GEMMs
constexpr int CHUNK = 30;       // 4080 = 136*30
constexpr int NCHUNK = NH / CHUNK;

// ---------------------------------------------------------------------------
// Zero the psi2 accumulator (harness poisons workspace; atomics need zeros)
// ---------------------------------------------------------------------------
__global__ void k_zero(float* __restrict__ p, int n) {
  int i = blockIdx.x * blockDim.x + threadIdx.x;
  if (i < n) p[i] = 0.f;
}

// ---------------------------------------------------------------------------
// Per-(n,d) prep: Hankel gather + reciprocals + per-n log-scalars
//   LHSb[n, 0:256]=w*Xm, LHSb[n,256:512]=w   (w = 1/(2Xv+l2))
//   LHSp[n, 0:256]=r*Xm, LHSp[n,256:512]=r   (r = 1/(Xv+l2))
//   lc1[n] = -0.5*sum(log1p(Xv/l2) + Xm^2*r)          (psi1 log-prefactor)
//   lc2[n] = -0.5*sum(log1p(2Xv/l2) + 2*w*Xm^2)       (psi2 log-prefactor)
// ---------------------------------------------------------------------------
__global__ __launch_bounds__(256) void k_prep(
    const float* __restrict__ Xm_m, const float* __restrict__ Xm_v,
    const float* __restrict__ ls,
    float* __restrict__ LHSb, float* __restrict__ LHSp,
    float* __restrict__ lc1, float* __restrict__ lc2) {
  int n = blockIdx.x;
  int d = threadIdx.x;                  // 0..255
  int row = 1 + n + (d >> 3);
  int q = d & 7;
  float xm = Xm_m[row * Q_ + q];
  float xv = Xm_v[row * Q_ + q];
  float l = ls[d];
  float l2 = l * l;
  float r = 1.f / (xv + l2);
  float w = 1.f / (2.f * xv + l2);
  LHSb[n * K2 + d]       = w * xm;
  LHSb[n * K2 + 256 + d] = w;
  LHSp[n * K2 + d]       = r * xm;
  LHSp[n * K2 + 256 + d] = r;
  float a1 = log1pf(xv / l2) + xm * xm * r;
  float a2 = log1pf(2.f * xv / l2) + 2.f * w * xm * xm;
  __shared__ float s1[256], s2[256];
  s1[d] = a1; s2[d] = a2;
  __syncthreads();
  for (int s = 128; s > 0; s >>= 1) {
    if (d < s) { s1[d] += s1[d + s]; s2[d] += s2[d + s]; }
    __syncthreads();
  }
  if (d == 0) { lc1[n] = -0.5f * s1[0]; lc2[n] = -0.5f * s2[0]; }
}

// ---------------------------------------------------------------------------
// RHS builder: RHSb=[Z, -0.25 Z^2], RHSp=[Z, -0.5 Z^2], Zs=Z/l
// ---------------------------------------------------------------------------
__global__ __launch_bounds__(256) void k_rhs(
    const float* __restrict__ Z, const float* __restrict__ ls,
    float* __restrict__ RHSb, float* __restrict__ RHSp, float* __restrict__ Zs) {
  int m = blockIdx.x;
  int d = threadIdx.x;
  float z = Z[m * DIN + d];
  RHSb[m * K2 + d]       = z;
  RHSb[m * K2 + 256 + d] = -0.25f * z * z;
  RHSp[m * K2 + d]       = z;
  RHSp[m * K2 + 256 + d] = -0.5f * z * z;
  Zs[m * DIN + d] = z / ls[d];
}

// ---------------------------------------------------------------------------
// Generic K=512 WMMA f32 GEMM: OUT[NH x 128] = LHS[NH x 512] * RHS[128 x 512]^T
// One block = one 16-row n-tile; 8 waves cover the 8 m-tiles. LHS tile staged
// in LDS (conflict-free pad), RHS streamed from L2.
// ---------------------------------------------------------------------------
__global__ __launch_bounds__(256) void k_gemm512(
    const float* __restrict__ LHS, const float* __restrict__ RHS,
    float* __restrict__ OUT) {
  extern __shared__ float sm[];         // 16 x (K2+4)
  const int PADK = K2 + 4;
  int nT = blockIdx.x;
  int t = threadIdx.x, wv = t >> 5, l = t & 31, r = l & 15, hi = l >> 4;
  for (int e = t; e < 16 * K2; e += 256) {
    int rr = e >> 9, kk = e & (K2 - 1);
    sm[rr * PADK + kk] = LHS[(nT * 16 + rr) * K2 + kk];
  }
  __syncthreads();
  const float* Arow = sm + r * PADK + 2 * hi;                 // A: 16x4 tile rows
  const float* Brow = RHS + (wv * 16 + r) * K2 + 2 * hi;      // B: 4x16 (cols = RHS rows)
  v8f acc = {};
  #pragma unroll 16
  for (int kk = 0; kk < K2; kk += 4) {
    v2f a = *(const v2f*)(Arow + kk);
    v2f b = *(const v2f*)(Brow + kk);
    acc = __builtin_amdgcn_wmma_f32_16x16x4_f32(false, a, false, b,
                                                (short)0, acc, false, false);
  }
  #pragma unroll
  for (int j = 0; j < 8; ++j)
    OUT[(nT * 16 + j + 8 * hi) * M_ + wv * 16 + r] = acc[j];
}

// ---------------------------------------------------------------------------
// E[n,m] = exp(0.5*lc2[n] + OUTb[n,m])   (in place over OUTb)
// ---------------------------------------------------------------------------
__global__ __launch_bounds__(256) void k_expE(float* __restrict__ Eb,
                                              const float* __restrict__ lc2) {
  int idx = blockIdx.x * 256 + threadIdx.x;   // NH*M_ total, exact multiple
  int n = idx >> 7;
  Eb[idx] = __expf(0.5f * lc2[n] + Eb[idx]);
}

// ---------------------------------------------------------------------------
// P1Y[m,q] = sum_n psi1[n,m]*Y[n,q],  psi1 = kv*exp(lc1[n] + OUTp[n,m])
// ---------------------------------------------------------------------------
__global__ __launch_bounds__(256) void k_p1y(
    const float* __restrict__ OUTp, const float* __restrict__ lc1,
    const float* __restrict__ Y, const float* __restrict__ kern_var,
    float* __restrict__ P1Y) {
  __shared__ float red[256 * Q_];
  int m = blockIdx.x, t = threadIdx.x;
  float kv = *kern_var;
  float acc[Q_];
  #pragma unroll
  for (int q = 0; q < Q_; ++q) acc[q] = 0.f;
  for (int n = t; n < NH; n += 256) {
    float p = kv * __expf(lc1[n] + OUTp[n * M_ + m]);
    #pragma unroll
    for (int q = 0; q < Q_; ++q) acc[q] += p * Y[n * Q_ + q];
  }
  #pragma unroll
  for (int q = 0; q < Q_; ++q) red[t * Q_ + q] = acc[q];
  __syncthreads();
  for (int s = 128; s > 0; s >>= 1) {
    if (t < s)
      #pragma unroll
      for (int q = 0; q < Q_; ++q) red[t * Q_ + q] += red[(t + s) * Q_ + q];
    __syncthreads();
  }
  if (t < Q_) P1Y[m * Q_ + t] = red[t];
}

// ---------------------------------------------------------------------------
// psi2 core: S[m,p] += sum_n E[n,m]*E[n,p]*exp(-0.5*G[n,m,p])
// G_n = (Z * diag(w_n)) @ Z^T via V_WMMA_F32_16X16X4_F32.
// Block = (mT, n-chunk); 8 waves = 8 pT tiles. Z + scaled A-tile + E row in LDS.
// ---------------------------------------------------------------------------
__global__ __launch_bounds__(256) void k_psi2(
    const float* __restrict__ LHSb,   // w_n at LHSb + n*K2 + 256
    const float* __restrict__ E,      // [NH, 128]
    const float* __restrict__ Z,      // [128, 256]
    float* __restrict__ S) {          // [128, 128] f32 accumulator
  extern __shared__ float sm[];
  const int PADW = DIN + 4;           // 260 -> 4-bank row stride, conflict-free
  float* Zl = sm;                     // 128*260
  float* Zw = Zl + M_ * PADW;         // 16*260 (w_n-scaled A tile)
  float* En = Zw + 16 * PADW;         // 128
  float* wn = En + M_;                // 256

  int t = threadIdx.x, wv = t >> 5, l = t & 31, r = l & 15, hi = l >> 4;
  int mT = blockIdx.x;                // 0..7
  int c0 = blockIdx.y * CHUNK;

  for (int e = t; e < M_ * DIN; e += 256) {
    int rr = e >> 8, kk = e & (DIN - 1);
    Zl[rr * PADW + kk] = Z[e];
  }

  float s_acc[8];
  #pragma unroll
  for (int j = 0; j < 8; ++j) s_acc[j] = 0.f;

  const float* Arow = Zw + r * PADW + 2 * hi;
  const float* Brow = Zl + (wv * 16 + r) * PADW + 2 * hi;
  int stage_r = t >> 4, stage_k = (t & 15) << 4;
  const float* zsrc = Zl + (mT * 16 + stage_r) * PADW + stage_k;
  float* zdst = Zw + stage_r * PADW + stage_k;
  int ep_idx = wv * 16 + r;

  for (int i = 0; i < CHUNK; ++i) {
    int n = c0 + i;
    __syncthreads();                          // prior iter's reads complete
    wn[t] = LHSb[n * K2 + 256 + t];
    if (t < M_) En[t] = E[n * M_ + t];
    __syncthreads();
    #pragma unroll
    for (int u = 0; u < 16; ++u) zdst[u] = zsrc[u] * wn[stage_k + u];
    __syncthreads();

    v8f acc = {};
    #pragma unroll
    for (int kk = 0; kk < DIN; kk += 4) {
      v2f a = *(const v2f*)(Arow + kk);
      v2f b = *(const v2f*)(Brow + kk);
      acc = __builtin_amdgcn_wmma_f32_16x16x4_f32(false, a, false, b,
                                                  (short)0, acc, false, false);
    }
    float ep = En[ep_idx];
    #pragma unroll
    for (int j = 0; j < 8; ++j) {
      float em = En[mT * 16 + j + 8 * hi];
      s_acc[j] += ep * em * __expf(-0.5f * acc[j]);
    }
  }
  #pragma unroll
  for (int j = 0; j < 8; ++j)
    atomicAdd(&S[(mT * 16 + j + 8 * hi) * M_ + wv * 16 + r], s_acc[j]);
}

// ---------------------------------------------------------------------------
// Final collapsed-ELBO solver: one 128-thread block, 192KB LDS.
// ---------------------------------------------------------------------------
__device__ __forceinline__ void chol128(float* A, int t) {
  for (int k = 0; k < M_; ++k) {
    __syncthreads();
    float dk = sqrtf(A[k * M_ + k]);
    __syncthreads();
    if (t >= k) A[t * M_ + k] = (t == k) ? dk : (A[t * M_ + k] / dk);
    __syncthreads();
    if (t > k) {
      float lik = A[t * M_ + k];
      for (int j = k + 1; j <= t; ++j) A[t * M_ + j] -= lik * A[j * M_ + k];
    }
  }
  __syncthreads();
}

__device__ __forceinline__ float reduce128(volatile float* red, float v, int t) {
  red[t] = v;
  __syncthreads();
  for (int s = 64; s > 0; s >>= 1) {
    if (t < s) red[t] += red[t + s];
    __syncthreads();
  }
  float r = red[0];
  __syncthreads();
  return r;
}

__global__ __launch_bounds__(128) void k_final(
    const float* __restrict__ Zs, const float* __restrict__ Sbuf,
    const float* __restrict__ P1Y, const float* __restrict__ Y,
    const float* __restrict__ kern_var, const float* __restrict__ lik_var,
    float* __restrict__ out) {
  extern __shared__ float sm[];
  float* Lm  = sm;                 // Kuu -> L
  float* Bm  = Lm + M_ * M_;       // psi2 -> AAT -> B -> LB
  float* Tm  = Bm + M_ * M_;       // L^{-1} psi2 ; later y1/y2
  float* red = Tm + M_ * M_;       // 128
  int c = threadIdx.x;             // column owner
  float kv = *kern_var, s2 = *lik_var;

  // Kuu and psi2 (ZQ = 0.25*d2 with d2 in lengthscale-normalized coords)
  for (int i = 0; i < M_; ++i) {
    const float* zi = Zs + i * DIN;
    const float* zj = Zs + c * DIN;
    float d2 = 0.f;
    for (int d = 0; d < DIN; ++d) { float df = zi[d] - zj[d]; d2 += df * df; }
    Lm[i * M_ + c] = kv * __expf(-0.5f * d2) + ((i == c) ? 1e-6f : 0.f);
    Bm[i * M_ + c] = kv * kv * __expf(-0.25f * d2) * Sbuf[i * M_ + c];
  }
  __syncthreads();

  chol128(Lm, c);                                   // L = chol(Kuu)

  // Tm = L^{-1} psi2 (each thread: forward solve of own column)
  for (int i = 0; i < M_; ++i) {
    float x = Bm[i * M_ + c];
    for (int k = 0; k < i; ++k) x -= Lm[i * M_ + k] * Tm[k * M_ + c];
    Tm[i * M_ + c] = x / Lm[i * M_ + i];
  }
  __syncthreads();

  // AAT = L^{-1} Tm^T / s2  -> Bm
  for (int i = 0; i < M_; ++i) {
    float x = Tm[c * M_ + i];
    for (int k = 0; k < i; ++k) x -= Lm[i * M_ + k] * Bm[k * M_ + c];
    Bm[i * M_ + c] = x / Lm[i * M_ + i];
  }
  __syncthreads();
  float inv_s2 = 1.f / s2;
  for (int i = 0; i < M_; ++i) Bm[i * M_ + c] *= inv_s2;
  float trAAT = reduce128(red, Bm[c * M_ + c], c);
  Bm[c * M_ + c] += 1.f;                            // B = AAT + I
  __syncthreads();

  chol128(Bm, c);                                   // LB = chol(B)
  float logdetB = 2.f * reduce128(red, __logf(Bm[c * M_ + c]), c);

  // y1 = L^{-1} P1Y ; y2 = LB^{-1} y1 ; c_mat = y2 / s2^2
  float* y1 = Tm;               // [M_, Q_]
  float* y2 = Tm + M_ * Q_;     // [M_, Q_]
  if (c < Q_) {
    for (int i = 0; i < M_; ++i) {
      float x = P1Y[i * Q_ + c];
      for (int k = 0; k < i; ++k) x -= Lm[i * M_ + k] * y1[k * Q_ + c];
      y1[i * Q_ + c] = x / Lm[i * M_ + i];
    }
    for (int i = 0; i < M_; ++i) {
      float x = y1[i * Q_ + c];
      for (int k = 0; k < i; ++k) x -= Bm[i * M_ + k] * y2[k * Q_ + c];
      y2[i * Q_ + c] = x / Bm[i * M_ + i];
    }
  }
  __syncthreads();
  float cs = 0.f;
  if (c < Q_)
    for (int i = 0; i < M_; ++i) { float v = y2[i * Q_ + c]; cs += v * v; }
  float sumY2sq = reduce128(red, cs, c);
  float sumC2 = sumY2sq / (s2 * s2);                // (y2/s2^2)^2 summed * s2^2 ... = sum c^2 * s2^2? no:
  // c = y2 / s2^2  -> sum c^2 = sum y2^2 / s2^4 ; but A@Y already had /s, c /s:
  // c = solve(LB, solve(L,P1Y))/s^2  => sum c^2 = sumY2sq / s2^2
  sumC2 = sumY2sq / (s2 * s2);

  float sy = 0.f;
  for (int idx = c; idx < NH * Q_; idx += M_) { float v = Y[idx]; sy += v * v; }
  float sumYY = reduce128(red, sy, c);

  if (c == 0) {
    float psi0 = (float)NH * kv;
    float two_pi = 6.283185307179586f;
    float bound = -0.5f * (float)(NH * Q_) * __logf(two_pi * s2);
    bound += -0.5f / s2 * sumYY;
    bound += -0.5f * (float)Q_ * (psi0 / s2 - trAAT);
    bound += -0.5f * (float)Q_ * logdetB;
    bound += 0.5f * sumC2;
    out[0] = bound;
  }
}

// ---------------------------------------------------------------------------
// Host-side launch
// ---------------------------------------------------------------------------
extern "C" void kernel_launch(void* const* d_in, const int* in_sizes, int n_in,
                              void* d_out, int out_size, void* d_ws, size_t ws_size,
                              hipStream_t stream) {
  const float* Xm_m = (const float*)d_in[0];
  const float* Xm_v = (const float*)d_in[1];
  const float* Z    = (const float*)d_in[2];
  const float* Y    = (const float*)d_in[3];
  const float* kv   = (const float*)d_in[4];
  const float* ls   = (const float*)d_in[5];
  const float* lv   = (const float*)d_in[6];
  (void)in_sizes; (void)n_in; (void)out_size; (void)ws_size;

  float* ws = (float*)d_ws;
  float* LHSb = ws;                 ws += (size_t)NH * K2;
  float* LHSp = ws;                 ws += (size_t)NH * K2;
  float* RHSb = ws;                 ws += (size_t)M_ * K2;
  float* RHSp = ws;                 ws += (size_t)M_ * K2;
  float* OUTb = ws;                 ws += (size_t)NH * M_;   // becomes E in place
  float* OUTp = ws;                 ws += (size_t)NH * M_;
  float* Zs   = ws;                 ws += (size_t)M_ * DIN;
  float* lc1  = ws;                 ws += 4096;
  float* lc2  = ws;                 ws += 4096;
  float* Sbuf = ws;                 ws += (size_t)M_ * M_;
  float* P1Y  = ws;                 ws += (size_t)M_ * Q_;

  k_zero<<<(M_ * M_ + 255) / 256, 256, 0, stream>>>(Sbuf, M_ * M_);
  k_prep<<<NH, 256, 0, stream>>>(Xm_m, Xm_v, ls, LHSb, LHSp, lc1, lc2);
  k_rhs<<<M_, 256, 0, stream>>>(Z, ls, RHSb, RHSp, Zs);

  size_t sh_gemm = (size_t)16 * (K2 + 4) * sizeof(float);           // 33 KB
  k_gemm512<<<NH / 16, 256, sh_gemm, stream>>>(LHSb, RHSb, OUTb);
  k_gemm512<<<NH / 16, 256, sh_gemm, stream>>>(LHSp, RHSp, OUTp);

  k_expE<<<(NH * M_) / 256, 256, 0, stream>>>(OUTb, lc2);
  k_p1y<<<M_, 256, 0, stream>>>(OUTp, lc1, Y, kv, P1Y);

  size_t sh_psi2 = (size_t)(M_ * (DIN + 4) + 16 * (DIN + 4) + M_ + DIN) * sizeof(float);
  k_psi2<<<dim3(8, NCHUNK), 256, sh_psi2, stream>>>(LHSb, OUTb, Z, Sbuf);

  size_t sh_fin = (size_t)(3 * M_ * M_ + 128) * sizeof(float);      // 192 KB
  k_final<<<1, 128, sh_fin, stream>>>(Zs, Sbuf, P1Y, Y, kv, lv, (float*)d_out);
}